// StylometricLoss_70368744178410
// MI455X (gfx1250) — compile-verified
//
#include <hip/hip_runtime.h>

typedef float f32x2 __attribute__((ext_vector_type(2)));
typedef float f32x4 __attribute__((ext_vector_type(4)));
typedef float f32x8 __attribute__((ext_vector_type(8)));

#define NBINS 25
#define MAXLEN 80.0f
#define EPSF 1e-8f
#define BB 4
#define LL 1024
#define VV 50257
#define NPART 8192   // entropy partial-sum blocks

// ---------------------------------------------------------------------------
// Kernel 1: grand sum of p*log(p+eps) over the flat tensor (bandwidth bound:
// 823 MB read once -> ~35 us at 23.3 TB/s).  Flat length B*L*V is divisible
// by 4 -> aligned float4 (global_load_b128) stream, non-temporal (no reuse,
// stream > 192 MB L2).  Two independent loads in flight per iteration.
// Deterministic: one partial per block, reduced in fixed order later.
// ---------------------------------------------------------------------------
__global__ void entropy_sum_kernel(const float* __restrict__ p,
                                   float* __restrict__ partials, long n4) {
  const f32x4* p4 = (const f32x4*)p;
  const long stride = (long)gridDim.x * blockDim.x;
  long i = (long)blockIdx.x * blockDim.x + threadIdx.x;
  long j = i + stride;
  float acc = 0.0f, accB = 0.0f;
  for (; j < n4; i += 2 * stride, j += 2 * stride) {
    f32x4 v = __builtin_nontemporal_load(&p4[i]);
    f32x4 w = __builtin_nontemporal_load(&p4[j]);
    acc  += v.x * __logf(v.x + EPSF);
    acc  += v.y * __logf(v.y + EPSF);
    acc  += v.z * __logf(v.z + EPSF);
    acc  += v.w * __logf(v.w + EPSF);
    accB += w.x * __logf(w.x + EPSF);
    accB += w.y * __logf(w.y + EPSF);
    accB += w.z * __logf(w.z + EPSF);
    accB += w.w * __logf(w.w + EPSF);
  }
  if (i < n4) {
    f32x4 v = __builtin_nontemporal_load(&p4[i]);
    acc += v.x * __logf(v.x + EPSF);
    acc += v.y * __logf(v.y + EPSF);
    acc += v.z * __logf(v.z + EPSF);
    acc += v.w * __logf(v.w + EPSF);
  }
  acc += accB;
  // wave32 reduction
  for (int off = 16; off > 0; off >>= 1) acc += __shfl_down(acc, off, 32);
  __shared__ float wpart[8];
  const int lane = threadIdx.x & 31, w = threadIdx.x >> 5;
  if (lane == 0) wpart[w] = acc;
  __syncthreads();
  if (threadIdx.x == 0) {
    float s = 0.0f;
    for (int k = 0; k < 8; ++k) s += wpart[k];
    partials[blockIdx.x] = s;   // no atomics: deterministic
  }
}

// ---------------------------------------------------------------------------
// Kernel 2: p_boundary[b,t] = sum of 5 gathered vocab columns (tiny).
// ---------------------------------------------------------------------------
__global__ void boundary_kernel(const float* __restrict__ p,
                                const int* __restrict__ ids, int nids,
                                float* __restrict__ pb) {
  const int idx = blockIdx.x * blockDim.x + threadIdx.x;
  if (idx >= BB * LL) return;
  const float* row = p + (long)idx * VV;
  float s = 0.0f;
  for (int k = 0; k < nids; ++k) s += row[ids[k]];
  pb[idx] = s;
}

// ---------------------------------------------------------------------------
// Kernel 3: counter scan + WMMA soft-histogram + symmetric KL + final loss.
// hist[b,bin] = sum_t pb[b,t] * g_norm(c[b,t], bin) is a (1xT)(Tx25) matvec
// -> V_WMMA_F32_16X16X4_F32 accumulation chains (2 per wave for 25 bins).
// A(16x4) f32 layout: lanes 0-15 hold K=0 (vgpr0), K=1 (vgpr1); lanes 16-31
// hold K=2,3.  B(4x16): lanes 0-15 = rows K=0/1 of col n, lanes 16-31 = K=2/3.
// Each wave writes its own histogram partial row (no FP atomics anywhere).
// ---------------------------------------------------------------------------
__global__ void __launch_bounds__(256, 1)
hist_loss_kernel(const float* __restrict__ pb_g, const float* __restrict__ human,
                 const float* __restrict__ target, const float* __restrict__ ent_part,
                 float* __restrict__ out) {
  __shared__ float pb[BB * LL];       // boundary probs
  __shared__ float cnt[BB * LL];      // counter value at each step (pre-update)
  __shared__ float rnorm[BB * LL];    // 1 / (sum_bins g + eps)
  __shared__ float histpart[8][32];   // per-wave histogram partials
  __shared__ float wsum[8];
  __shared__ float ent_total;
  const int tid = threadIdx.x;
  const int wave = tid >> 5, lane = tid & 31;

  // stage boundary probs into LDS
  for (int i = tid; i < BB * LL; i += 256) pb[i] = pb_g[i];

  // deterministic fixed-order reduction of the 8192 entropy partials
  {
    float es = 0.0f;
    for (int i = tid; i < NPART; i += 256) es += ent_part[i];
    for (int off = 16; off > 0; off >>= 1) es += __shfl_down(es, off, 32);
    if (lane == 0) wsum[wave] = es;
  }
  __syncthreads();
  if (tid == 0) {
    float s = 0.0f;
    for (int k = 0; k < 8; ++k) s += wsum[k];
    ent_total = s;
  }

  // serial counter recurrence: c <- c*(1-pb)+1 (independent of hist)
  if (tid < BB) {
    float c = 1.0f;
    const float* pbr = &pb[tid * LL];
    float* cr = &cnt[tid * LL];
    for (int t = 0; t < LL; ++t) {
      cr[t] = c;
      c = c * (1.0f - pbr[t]) + 1.0f;
    }
  }
  __syncthreads();

  const float inv_sigma = 1.0f / (MAXLEN / (float)NBINS + EPSF); // sigma = 3.2
  const float binw = MAXLEN / (float)(NBINS - 1);                // linspace step

  // per-(b,t) Gaussian normalizers, fully parallel (256 threads x 16 each)
  for (int i = tid; i < BB * LL; i += 256) {
    const float c = cnt[i];
    float s = 0.0f;
    for (int bin = 0; bin < NBINS; ++bin) {
      const float d = (c - bin * binw) * inv_sigma;
      s += __expf(-0.5f * d * d);
    }
    rnorm[i] = 1.0f / (s + EPSF);
  }
  __syncthreads();

  // WMMA phase: wave w handles batch w>>1, half w&1 (512 t in chunks of 4)
  const int b = wave >> 1, half = wave & 1;
  const int n = lane & 15;
  const int kbase = (lane < 16) ? 0 : 2;
  const float cen_lo = n * binw;
  const float cen_hi = (16 + n) * binw;
  const bool hi_valid = (16 + n) < NBINS;
  const int base = b * LL + half * (LL / 2);
  f32x8 acc0 = {}, acc1 = {};
  for (int chunk = 0; chunk < (LL / 2) / 4; ++chunk) {
    const int t0 = base + chunk * 4;
    f32x2 A;
    A.x = pb[t0 + kbase];
    A.y = pb[t0 + kbase + 1];
    const float c0 = cnt[t0 + kbase], c1 = cnt[t0 + kbase + 1];
    const float r0 = rnorm[t0 + kbase], r1 = rnorm[t0 + kbase + 1];
    f32x2 B0, B1;
    float d;
    d = (c0 - cen_lo) * inv_sigma; B0.x = __expf(-0.5f * d * d) * r0;
    d = (c1 - cen_lo) * inv_sigma; B0.y = __expf(-0.5f * d * d) * r1;
    d = (c0 - cen_hi) * inv_sigma; B1.x = hi_valid ? __expf(-0.5f * d * d) * r0 : 0.0f;
    d = (c1 - cen_hi) * inv_sigma; B1.y = hi_valid ? __expf(-0.5f * d * d) * r1 : 0.0f;
    acc0 = __builtin_amdgcn_wmma_f32_16x16x4_f32(false, A, false, B0,
                                                 (short)0, acc0, false, false);
    acc1 = __builtin_amdgcn_wmma_f32_16x16x4_f32(false, A, false, B1,
                                                 (short)0, acc1, false, false);
  }
  // D row M=0 lives in vgpr0 of lanes 0-15 (N = lane); plain stores per wave
  if (lane < 16) {
    histpart[wave][n] = acc0[0];
    if (n < NBINS - 16) histpart[wave][16 + n] = acc1[0];
  }
  __syncthreads();

  if (tid == 0) {
    float klsum = 0.0f;
    for (int bi = 0; bi < BB; ++bi) {
      float h[NBINS], tot = 0.0f;
      for (int i = 0; i < NBINS; ++i) {
        h[i] = histpart[2 * bi][i] + histpart[2 * bi + 1][i] + EPSF;
        tot += h[i];
      }
      float kl_qp = 0.0f, kl_pq = 0.0f;
      for (int i = 0; i < NBINS; ++i) {
        const float Q = h[i] / tot + EPSF;
        const float P = human[i] + EPSF;
        kl_qp += Q * __logf(Q / P);
        kl_pq += P * __logf(P / Q);
      }
      klsum += 0.5f * (kl_qp + kl_pq);
    }
    const float kl_burst = klsum / (float)BB;
    const float ent_mean = -ent_total / (float)(BB * LL);
    const float dlt = ent_mean - target[0];
    out[0] = 0.7f * kl_burst + 0.3f * dlt * dlt;
  }
}

extern "C" void kernel_launch(void* const* d_in, const int* in_sizes, int n_in,
                              void* d_out, int out_size, void* d_ws, size_t ws_size,
                              hipStream_t stream) {
  const float* prob   = (const float*)d_in[0];  // (4,1024,50257) f32
  const float* human  = (const float*)d_in[1];  // (25,) f32
  const float* target = (const float*)d_in[2];  // (1,)  f32
  const int*   ids    = (const int*)d_in[3];    // (5,)  i32
  float* out      = (float*)d_out;
  float* pb       = (float*)d_ws;                          // 4096 floats
  float* ent_part = (float*)((char*)d_ws + BB * LL * 4);   // 8192 floats

  const long n4 = (long)BB * LL * VV / 4;  // divisible exactly
  entropy_sum_kernel<<<NPART, 256, 0, stream>>>(prob, ent_part, n4);
  boundary_kernel<<<(BB * LL + 255) / 256, 256, 0, stream>>>(prob, ids, in_sizes[3], pb);
  hist_loss_kernel<<<1, 256, 0, stream>>>(pb, human, target, ent_part, out);
}